// SimpleET_35192962023785
// MI455X (gfx1250) — compile-verified
//
#include <hip/hip_runtime.h>

typedef float v2f __attribute__((ext_vector_type(2)));
typedef float v8f __attribute__((ext_vector_type(8)));

#define N_ATOMS 2048
#define BATCH 2
#define RECV_PER_BLOCK 8        // 8 waves of 32 -> 256 threads, one receiver per wave
#define CUTOFF_SQ 0.49f         // (0.7 nm)^2  -- dist<0.7  <=>  d2<0.49
#define MAX_DISP 0.02f

__device__ __forceinline__ float fast_rcp(float x)  { return __builtin_amdgcn_rcpf(x); }
__device__ __forceinline__ float fast_rsq(float x)  { return __builtin_amdgcn_rsqf(x); }

__device__ __forceinline__ float silu_f(float x) {
    // x * sigmoid(x); exp->0 or inf limits are exact with rcp
    return x * fast_rcp(1.0f + __expf(-x));
}
__device__ __forceinline__ float tanh_f(float x) {
    float xc = fminf(fmaxf(x, -15.0f), 15.0f);     // avoid inf*0 in rcp form
    float e  = __expf(2.0f * xc);
    return (e - 1.0f) * fast_rcp(e + 1.0f);
}

__global__ __launch_bounds__(256) void simple_et_kernel(
    const float* __restrict__ x_nm, const int* __restrict__ atom_types,
    const int* __restrict__ mask_in, const float* __restrict__ emb_W,
    const float* __restrict__ msg_w1, const float* __restrict__ msg_b1,
    const float* __restrict__ msg_w2, const float* __restrict__ msg_b2,
    const float* __restrict__ upd_w1, const float* __restrict__ upd_b1,
    const float* __restrict__ upd_w2, const float* __restrict__ upd_b2,
    const float* __restrict__ dx_w, const float* __restrict__ dx_b,
    const float* __restrict__ dv_w, const float* __restrict__ dv_b,
    float* __restrict__ out)
{
    // node n: snode[2n] = {x0,x1,x2,mask}, snode[2n+1] = {h0,h1,h2,0}   (64 KB)
    __shared__ float4 snode[N_ATOMS * 2];

    const int tid  = threadIdx.x;
    const int b    = blockIdx.x / (N_ATOMS / RECV_PER_BLOCK);
    const int row0 = (blockIdx.x % (N_ATOMS / RECV_PER_BLOCK)) * RECV_PER_BLOCK;

    // ---- stage all senders of this batch into LDS ----
    for (int n = tid; n < N_ATOMS; n += 256) {
        int t = atom_types[b * N_ATOMS + n];
        float mv = mask_in[b * N_ATOMS + n] ? 1.0f : 0.0f;
        snode[2 * n]     = make_float4(x_nm[(b * N_ATOMS + n) * 3 + 0],
                                       x_nm[(b * N_ATOMS + n) * 3 + 1],
                                       x_nm[(b * N_ATOMS + n) * 3 + 2], mv);
        snode[2 * n + 1] = make_float4(emb_W[t * 3 + 0],
                                       emb_W[t * 3 + 1],
                                       emb_W[t * 3 + 2], 0.0f);
    }
    __syncthreads();

    const int lane = tid & 31;
    const int wave = tid >> 5;
    const int i    = row0 + wave;                 // receiver owned by this wave
    const int m    = lane & 15;                   // pair slot / A-row index
    const bool lo  = (lane < 16);

    const float4 xiq = snode[2 * i];
    const float  xi0 = xiq.x, xi1 = xiq.y, xi2 = xiq.z, maski = xiq.w;

    // ---- A1 = msg_w1^T padded to 16x4, with row 10 duplicating row 2 so that
    //      D1[2]'s upper lanes natively hold feature-2 (no cross-lane shuffle).
    // A layout: lanes 0-15 -> M=lane, K0/K1; lanes 16-31 -> M=lane-16, K2/K3.
    const int  fm = (m < 3) ? m : 2;                         // clamped feature idx
    const float fz = (m < 3 || m == 10) ? 1.0f : 0.0f;       // rows 0,1,2,10 live
    v2f A1, A2;
    A1[0] = fz * (lo ? msg_w1[0 * 3 + fm] : msg_w1[2 * 3 + fm]);
    A1[1] = fz * (lo ? msg_w1[1 * 3 + fm] : msg_w1[3 * 3 + fm]);
    // A2 = msg_w2^T (3x3) padded; only rows 0-2; K3 row is zero
    const float z3 = (m < 3) ? 1.0f : 0.0f;
    A2[0] = z3 * (lo ? msg_w2[0 * 3 + fm] : msg_w2[2 * 3 + fm]);
    A2[1] = (lo ? z3 * msg_w2[1 * 3 + fm] : 0.0f);

    // ---- bias accumulators (C/D layout: VGPR v = row v lanes 0-15, row v+8 lanes 16-31)
    v8f C1 = {}, C2 = {};
    C1[0] = lo ? msg_b1[0] : 0.0f;
    C1[1] = lo ? msg_b1[1] : 0.0f;
    C1[2] = msg_b1[2];                 // row 2 (lo) and duplicated row 10 (hi)
    C2[0] = lo ? msg_b2[0] : 0.0f;
    C2[1] = lo ? msg_b2[1] : 0.0f;
    C2[2] = lo ? msg_b2[2] : 0.0f;

    float acc0 = 0.0f, acc1 = 0.0f, acc2 = 0.0f;
    const float lmask = lo ? 1.0f : 0.0f;

#pragma unroll 2
    for (int jc = 0; jc < N_ATOMS; jc += 16) {
        const int j = jc + m;                      // this lane's sender
        const float4 xq = snode[2 * j];            // ds_load_b128
        const float4 hq = snode[2 * j + 1];        // ds_load_b128

        const float d0 = xi0 - xq.x, d1 = xi1 - xq.y, d2c = xi2 - xq.z;
        const float d2 = d0 * d0 + d1 * d1 + d2c * d2c + 1e-9f;
        const float inv = fast_rsq(d2);            // v_rsq_f32 : 1/dist
        const float validf = ((d2 < CUTOFF_SQ) && (j != i)) ? (xq.w * maski) : 0.0f;
        const float radial = validf * fminf(inv, 1.0e3f);          // 1/max(dist,1e-3)
        const float dscale = validf * lmask * fminf(inv, 1.0e6f);  // 1/max(dist,1e-6)
        const float dir0 = d0 * dscale, dir1 = d1 * dscale, dir2 = d2c * dscale;

        // B1 = msg_in^T (4 x 16 pairs): K0=h0,K1=h1 (lanes 0-15); K2=h2,K3=radial (lanes 16-31)
        v2f B1;
        B1[0] = lo ? hq.x : hq.z;
        B1[1] = lo ? hq.y : radial;

        // D1 = W1^T * msg_in^T + b1
        v8f D1 = __builtin_amdgcn_wmma_f32_16x16x4_f32(
            false, A1, false, B1, (short)0, C1, false, false);

        const float hd0 = silu_f(D1[0]);           // feature 0 (lower lanes)
        const float hd1 = silu_f(D1[1]);           // feature 1 (lower lanes)
        const float hd2 = silu_f(D1[2]);           // feature 2 on ALL lanes (row 2 / row 10)

        v2f B2;
        B2[0] = lo ? hd0 : hd2;                    // K0 | K2
        B2[1] = lo ? hd1 : 0.0f;                   // K1 | K3(=0)

        // D2 = W2^T * hidden^T + b2  -> msg channel c of pair n at VGPR c, lane n
        v8f D2 = __builtin_amdgcn_wmma_f32_16x16x4_f32(
            false, A2, false, B2, (short)0, C2, false, false);

        acc0 += D2[0] * dir0;
        acc1 += D2[1] * dir1;
        acc2 += D2[2] * dir2;
    }

    // reduce the 16 pair-lanes (upper half carries zeros)
    for (int off = 16; off > 0; off >>= 1) {
        acc0 += __shfl_xor(acc0, off, 32);
        acc1 += __shfl_xor(acc1, off, 32);
        acc2 += __shfl_xor(acc2, off, 32);
    }

    if (lane == 0) {
        const float v0 = acc0, v1 = acc1, v2 = acc2;
        const float4 hiq = snode[2 * i + 1];
        const float hi0 = hiq.x, hi1 = hiq.y, hi2 = hiq.z;

        const float u0 = silu_f(v0 * upd_w1[0] + v1 * upd_w1[3] + v2 * upd_w1[6] + upd_b1[0]);
        const float u1 = silu_f(v0 * upd_w1[1] + v1 * upd_w1[4] + v2 * upd_w1[7] + upd_b1[1]);
        const float u2 = silu_f(v0 * upd_w1[2] + v1 * upd_w1[5] + v2 * upd_w1[8] + upd_b1[2]);

        const float h0 = hi0 + u0 * upd_w2[0] + u1 * upd_w2[3] + u2 * upd_w2[6] + upd_b2[0];
        const float h1 = hi1 + u0 * upd_w2[1] + u1 * upd_w2[4] + u2 * upd_w2[7] + upd_b2[1];
        const float h2 = hi2 + u0 * upd_w2[2] + u1 * upd_w2[5] + u2 * upd_w2[8] + upd_b2[2];

        const int base = (b * N_ATOMS + i) * 3;
        const int half = BATCH * N_ATOMS * 3;
#pragma unroll
        for (int c = 0; c < 3; ++c) {
            const float px = h0 * dx_w[0 * 3 + c] + h1 * dx_w[1 * 3 + c] + h2 * dx_w[2 * 3 + c] + dx_b[c];
            const float pv = h0 * dv_w[0 * 3 + c] + h1 * dv_w[1 * 3 + c] + h2 * dv_w[2 * 3 + c] + dv_b[c];
            out[base + c]        = tanh_f(px) * MAX_DISP;
            out[half + base + c] = tanh_f(pv) * 1.0f;   // MAX_DISP / 0.02 == 1.0
        }
    }
}

extern "C" void kernel_launch(void* const* d_in, const int* in_sizes, int n_in,
                              void* d_out, int out_size, void* d_ws, size_t ws_size,
                              hipStream_t stream) {
    (void)in_sizes; (void)n_in; (void)d_ws; (void)ws_size; (void)out_size;
    const float* x_nm   = (const float*)d_in[0];
    const int*   types  = (const int*)  d_in[1];
    const int*   mask   = (const int*)  d_in[2];
    const float* emb_W  = (const float*)d_in[3];
    const float* msg_w1 = (const float*)d_in[4];
    const float* msg_b1 = (const float*)d_in[5];
    const float* msg_w2 = (const float*)d_in[6];
    const float* msg_b2 = (const float*)d_in[7];
    const float* upd_w1 = (const float*)d_in[8];
    const float* upd_b1 = (const float*)d_in[9];
    const float* upd_w2 = (const float*)d_in[10];
    const float* upd_b2 = (const float*)d_in[11];
    const float* dx_w   = (const float*)d_in[12];
    const float* dx_b   = (const float*)d_in[13];
    const float* dv_w   = (const float*)d_in[14];
    const float* dv_b   = (const float*)d_in[15];
    float* out = (float*)d_out;

    dim3 grid(BATCH * (N_ATOMS / RECV_PER_BLOCK));   // 512 blocks
    dim3 block(256);                                  // 8 wave32 per block
    simple_et_kernel<<<grid, block, 0, stream>>>(
        x_nm, types, mask, emb_W,
        msg_w1, msg_b1, msg_w2, msg_b2,
        upd_w1, upd_b1, upd_w2, upd_b2,
        dx_w, dx_b, dv_w, dv_b, out);
}